// GAT_74723841015999
// MI455X (gfx1250) — compile-verified
//
#include <hip/hip_runtime.h>
#include <cstdint>

// ---------------- problem constants (from reference setup_inputs) ------------
#define NN      50000     // nodes
#define F_IN    128
#define HEADS   8
#define HID     32
#define F1      256       // HEADS*HID
#define OUTC    64
#define E_RAW   800000
#define E_TOT   850000    // + self loops
#define NEG_SLOPE 0.2f
#define EPS_F   1e-16f

typedef float v2f __attribute__((ext_vector_type(2)));
typedef float v8f __attribute__((ext_vector_type(8)));

// Monotonic float<->int mapping so segment-max can use global_atomic_max_i32.
__device__ __forceinline__ int flip_f32(float f) {
  int i = __float_as_int(f);
  return i >= 0 ? i : (i ^ 0x7fffffff);
}
__device__ __forceinline__ float unflip_f32(int i) {
  return __int_as_float(i >= 0 ? i : (i ^ 0x7fffffff));
}

__device__ __forceinline__ void edge_sd(const int* __restrict__ ei, int e, int& s, int& d) {
  if (e < E_RAW) { s = ei[e]; d = ei[E_RAW + e]; }
  else           { s = e - E_RAW; d = s; }            // appended self loops
}

// ---------------- init kernels ----------------------------------------------
__global__ void init_f32(float* __restrict__ p, float v, int n) {
  int i = blockIdx.x * blockDim.x + threadIdx.x;
  if (i < n) p[i] = v;
}
__global__ void init_i32(int* __restrict__ p, int v, int n) {
  int i = blockIdx.x * blockDim.x + threadIdx.x;
  if (i < n) p[i] = v;
}

// ---------------- fp32 WMMA GEMM: C[MxN] = A[MxK] @ B[KxN], row-major --------
// One wave per 16x16 output tile; K stepped by 4 via v_wmma_f32_16x16x4_f32.
// N, K are compile-time so every unrolled A/B load is base + immediate IOFFSET
// (no per-iteration 64-bit address math), and the tile index math is shifts.
// Layouts (ISA 7.12.2): A 16x4: lanes0-15 -> {K,K+1}, lanes16-31 -> {K+2,K+3};
// B 4x16: v0 row K=half, v1 row K=2+half, N = lane&15;
// C/D: VGPR r -> row M = r + 8*half, col N = lane&15.
template <int N, int K>
__global__ void gemm_f32_wmma(const float* __restrict__ A, const float* __restrict__ B,
                              float* __restrict__ C, int M) {
  constexpr int NT = N >> 4;                                  // tiles along N (pow2)
  int wave = (blockIdx.x * blockDim.x + threadIdx.x) >> 5;    // wave32
  int lane = threadIdx.x & 31;
  int mt = wave / NT;                                         // constexpr -> shift
  if (mt >= (M >> 4)) return;                                 // wave-uniform exit
  int nt = wave & (NT - 1);
  int m0 = mt << 4, n0 = nt << 4;
  int half = lane >> 4;          // 0 or 1
  int l15  = lane & 15;

  const float* ap = A + (size_t)(m0 + l15) * K + half * 2;    // 8B aligned
  const float* bp = B + (size_t)half * N + n0 + l15;          // row k = half
  __builtin_prefetch(ap, 0, 3);                               // global_prefetch_b8 (near)

  v8f acc = {};
#pragma unroll
  for (int k = 0; k < K; k += 4) {
    v2f a = *(const v2f*)(ap + k);                            // A[m][k+2h .. k+2h+1]
    v2f b;
    b.x = bp[(size_t)k * N];                                  // B[k+half][n]
    b.y = bp[(size_t)(k + 2) * N];                            // B[k+2+half][n]
    acc = __builtin_amdgcn_wmma_f32_16x16x4_f32(
        /*neg_a=*/false, a, /*neg_b=*/false, b,
        /*c_mod=*/(short)0, acc, /*reuse_a=*/false, /*reuse_b=*/false);
  }
  float* crow = C + (size_t)(m0 + half * 8) * N + n0 + l15;
#pragma unroll
  for (int r = 0; r < 8; ++r) crow[(size_t)r * N] = acc[r];
}

// ---------------- attention score reduction: a[n,h] = <h[n,h,:], att[h,:]> ---
template <int H, int C>
__global__ void attn_scores(const float* __restrict__ h, const float* __restrict__ att_s,
                            const float* __restrict__ att_d, float* __restrict__ as_,
                            float* __restrict__ ad_, int n) {
  int i = blockIdx.x * blockDim.x + threadIdx.x;
  if (i >= n * H) return;
  int node = i / H, hd = i & (H - 1);                         // H pow2
  const float* hp = h + (size_t)node * H * C + (size_t)hd * C;
  const float* ws = att_s + hd * C;
  const float* wd = att_d + hd * C;
  float ss = 0.f, sd = 0.f;
#pragma unroll 8
  for (int c = 0; c < C; ++c) {
    float v = hp[c];
    ss = fmaf(v, ws[c], ss);
    sd = fmaf(v, wd[c], sd);
  }
  as_[i] = ss;
  ad_[i] = sd;
}

// ---------------- edge pass 1: segment max of leaky-relu logits --------------
template <int H>
__global__ void edge_max_k(const int* __restrict__ ei, const float* __restrict__ as_,
                           const float* __restrict__ ad_, int* __restrict__ m, int total) {
  int i = blockIdx.x * blockDim.x + threadIdx.x;
  if (i >= total) return;
  int e = i / H, hd = i & (H - 1);                            // H pow2 -> shifts
  int s, d; edge_sd(ei, e, s, d);
  float v = as_[s * H + hd] + ad_[d * H + hd];
  v = v > 0.f ? v : NEG_SLOPE * v;
  atomicMax(&m[d * H + hd], flip_f32(v));
}

// ---------------- edge pass 2: segment sum of exp(logit - max) ---------------
template <int H>
__global__ void edge_sum_k(const int* __restrict__ ei, const float* __restrict__ as_,
                           const float* __restrict__ ad_, const int* __restrict__ m,
                           float* __restrict__ ssum, int total) {
  int i = blockIdx.x * blockDim.x + threadIdx.x;
  if (i >= total) return;
  int e = i / H, hd = i & (H - 1);
  int s, d; edge_sd(ei, e, s, d);
  float v = as_[s * H + hd] + ad_[d * H + hd];
  v = v > 0.f ? v : NEG_SLOPE * v;
  float ex = expf(v - unflip_f32(m[d * H + hd]));
  atomicAdd(&ssum[d * H + hd], ex);
}

// ---------------- edge pass 3: out[dst] += alpha * h[src] --------------------
// Block covers EPB edges x (H*C) channels: gathers/scatters full contiguous rows.
template <int H, int C, int EPB>
__global__ void edge_aggregate(const int* __restrict__ ei, const float* __restrict__ as_,
                               const float* __restrict__ ad_, const int* __restrict__ m,
                               const float* __restrict__ ssum, const float* __restrict__ h,
                               float* __restrict__ out) {
  constexpr int HC = H * C;
  int e = blockIdx.x * EPB + (int)threadIdx.x / HC;
  if (e >= E_TOT) return;
  int t  = (int)threadIdx.x % HC;
  int hd = t / C;
  int s, d; edge_sd(ei, e, s, d);
  float v = as_[s * H + hd] + ad_[d * H + hd];
  v = v > 0.f ? v : NEG_SLOPE * v;
  float ex    = expf(v - unflip_f32(m[d * H + hd]));
  float alpha = ex / (ssum[d * H + hd] + EPS_F);
  atomicAdd(&out[(size_t)d * HC + t], h[(size_t)s * HC + t] * alpha);
}

// ---------------- epilogues --------------------------------------------------
__global__ void bias_elu(float* __restrict__ x, const float* __restrict__ b, int total, int fmask) {
  int i = blockIdx.x * blockDim.x + threadIdx.x;
  if (i >= total) return;
  float v = x[i] + b[i & fmask];
  x[i] = v > 0.f ? v : (expf(v) - 1.f);
}
__global__ void bias_add(float* __restrict__ x, const float* __restrict__ b, int total, int fmask) {
  int i = blockIdx.x * blockDim.x + threadIdx.x;
  if (i < total) x[i] = x[i] + b[i & fmask];
}

// ---------------- host side --------------------------------------------------
static inline int cdiv(long long a, long long b) { return (int)((a + b - 1) / b); }

extern "C" void kernel_launch(void* const* d_in, const int* in_sizes, int n_in,
                              void* d_out, int out_size, void* d_ws, size_t ws_size,
                              hipStream_t stream) {
  (void)in_sizes; (void)n_in; (void)out_size; (void)ws_size;

  const float* x        = (const float*)d_in[0];
  const int*   ei       = (const int*)d_in[1];     // [2, 800000] int32 (JAX x64 off)
  const float* W1       = (const float*)d_in[2];   // [128, 256]
  const float* att_src1 = (const float*)d_in[3];   // [8, 32]
  const float* att_dst1 = (const float*)d_in[4];
  const float* b1       = (const float*)d_in[5];   // [256]
  const float* W2       = (const float*)d_in[6];   // [256, 64]
  const float* att_src2 = (const float*)d_in[7];   // [1, 64]
  const float* att_dst2 = (const float*)d_in[8];
  const float* b2       = (const float*)d_in[9];   // [64]
  float*       out      = (float*)d_out;           // [50000, 64]

  // workspace layout (floats): h1 | x1 | as1 | ad1 | m1 | s1   (~109 MB)
  float* h1  = (float*)d_ws;                       // [NN, 256]
  float* x1  = h1 + (size_t)NN * F1;               // [NN, 256]
  float* as1 = x1 + (size_t)NN * F1;               // [NN, 8]
  float* ad1 = as1 + (size_t)NN * HEADS;
  int*   m1  = (int*)(ad1 + (size_t)NN * HEADS);
  float* s1  = (float*)(m1 + (size_t)NN * HEADS);
  // layer-2 reuse (h1 dead after x1 is built; logit buffers shrink to H=1)
  float* h2  = h1;        // [NN, 64]
  float* as2 = as1;  float* ad2 = ad1;
  int*   m2  = m1;   float* s2  = s1;

  const int T = 256;

  // ===== layer 1 =====
  {
    int tiles = (NN / 16) * (F1 / 16);
    gemm_f32_wmma<F1, F_IN><<<cdiv(tiles, 8), T, 0, stream>>>(x, W1, h1, NN);
  }
  attn_scores<HEADS, HID><<<cdiv((long long)NN * HEADS, T), T, 0, stream>>>(
      h1, att_src1, att_dst1, as1, ad1, NN);
  init_i32<<<cdiv((long long)NN * HEADS, T), T, 0, stream>>>(m1, (int)0x80000000, NN * HEADS);
  init_f32<<<cdiv((long long)NN * HEADS, T), T, 0, stream>>>(s1, 0.f, NN * HEADS);
  init_f32<<<cdiv((long long)NN * F1, T), T, 0, stream>>>(x1, 0.f, NN * F1);
  {
    int total = E_TOT * HEADS;
    edge_max_k<HEADS><<<cdiv(total, T), T, 0, stream>>>(ei, as1, ad1, m1, total);
    edge_sum_k<HEADS><<<cdiv(total, T), T, 0, stream>>>(ei, as1, ad1, m1, s1, total);
  }
  edge_aggregate<HEADS, HID, 1><<<E_TOT, T, 0, stream>>>(ei, as1, ad1, m1, s1, h1, x1);
  bias_elu<<<cdiv((long long)NN * F1, T), T, 0, stream>>>(x1, b1, NN * F1, F1 - 1);

  // ===== layer 2 (H=1, C=64; mean over 1 head == identity) =====
  {
    int tiles = (NN / 16) * (OUTC / 16);
    gemm_f32_wmma<OUTC, F1><<<cdiv(tiles, 8), T, 0, stream>>>(x1, W2, h2, NN);
  }
  attn_scores<1, OUTC><<<cdiv((long long)NN, T), T, 0, stream>>>(
      h2, att_src2, att_dst2, as2, ad2, NN);
  init_i32<<<cdiv((long long)NN, T), T, 0, stream>>>(m2, (int)0x80000000, NN);
  init_f32<<<cdiv((long long)NN, T), T, 0, stream>>>(s2, 0.f, NN);
  init_f32<<<cdiv((long long)NN * OUTC, T), T, 0, stream>>>(out, 0.f, NN * OUTC);
  edge_max_k<1><<<cdiv(E_TOT, T), T, 0, stream>>>(ei, as2, ad2, m2, E_TOT);
  edge_sum_k<1><<<cdiv(E_TOT, T), T, 0, stream>>>(ei, as2, ad2, m2, s2, E_TOT);
  edge_aggregate<1, OUTC, 4><<<cdiv(E_TOT, 4), T, 0, stream>>>(ei, as2, ad2, m2, s2, h2, out);
  bias_add<<<cdiv((long long)NN * OUTC, T), T, 0, stream>>>(out, b2, NN * OUTC, OUTC - 1);
}